// DigitCaps_51299089384102
// MI455X (gfx1250) — compile-verified
//
#include <hip/hip_runtime.h>
#include <math.h>

#define C_DIM 10
#define B_DIM 256
#define N_DIM 1152
#define DIN   8
#define DOUT  16

typedef __attribute__((ext_vector_type(2))) float v2f;
typedef __attribute__((ext_vector_type(4))) float v4f;
typedef __attribute__((ext_vector_type(8))) float v8f;

// d_out layout: [v_j | a_ij | u_hat] flat in reference return order
#define V_OFF  0
#define A_OFF  (C_DIM * B_DIM * DOUT)                 // 40960
#define UH_OFF (A_OFF + C_DIM * B_DIM * N_DIM)        // 2990080

#define NPW   9   // n-values per wave
#define WAVES 8   // waves per block

// ---------------------------------------------------------------------------
// Kernel A: u_hat[c,b,n,o] = sum_i u[b,n,i] * W[c,n,i,o] via f32 WMMA 16x16x4.
// One wave handles one (c, b-tile of 16, n) at a time: 16x8 @ 8x16 -> 16x16,
// i.e. two chained V_WMMA_F32_16X16X4_F32. Whole wave active (EXEC all 1s).
// ---------------------------------------------------------------------------
__global__ __launch_bounds__(256) void uhat_wmma_kernel(
    const float* __restrict__ u,   // (B,N,8)
    const float* __restrict__ W,   // (C,N,8,16)
    float* __restrict__ out)       // d_out base
{
    const int wave   = threadIdx.x >> 5;
    const int lane   = threadIdx.x & 31;
    const int laneHi = lane >> 4;        // 0: lanes 0-15, 1: lanes 16-31
    const int l15    = lane & 15;
    const int c      = blockIdx.x >> 4;
    const int b0     = (blockIdx.x & 15) << 4;
    int n            = blockIdx.y * (WAVES * NPW) + wave * NPW;

    const float* uRow = u + (size_t)(b0 + l15) * (N_DIM * DIN);
    float* uh = out + UH_OFF;

    for (int i = 0; i < NPW; ++i, ++n) {
        // A 16x4 frags (f32): lane row = M, VGPR0/1 = consecutive K, halves split K
        const float* ua = uRow + n * DIN + laneHi * 2;
        v2f a_lo = { ua[0], ua[1] };             // K 0..3
        v2f a_hi = { ua[4], ua[5] };             // K 4..7
        // B 4x16 frags: VGPR r holds row (K) r striped across lanes; hi half = K+2
        const float* wb = W + (((size_t)c * N_DIM + n) * DIN + laneHi * 2) * DOUT + l15;
        v2f b_lo = { wb[0],        wb[DOUT] };   // K 0..3
        v2f b_hi = { wb[4 * DOUT], wb[5 * DOUT] };// K 4..7

        v8f acc = {};
        acc = __builtin_amdgcn_wmma_f32_16x16x4_f32(
                  false, a_lo, false, b_lo, (short)0, acc, false, false);
        acc = __builtin_amdgcn_wmma_f32_16x16x4_f32(
                  false, a_hi, false, b_hi, (short)0, acc, false, false);

        // D 16x16: VGPR r -> rows r / r+8; column = lane&15 (coalesced b32 stores)
        size_t base = (((size_t)c * B_DIM + b0 + laneHi * 8) * N_DIM + n) * DOUT + l15;
        #pragma unroll
        for (int r = 0; r < 8; ++r)
            uh[base + (size_t)r * (N_DIM * DOUT)] = acc[r];
    }
}

// ---------------------------------------------------------------------------
// Kernel B: dynamic routing. One block per (c,b). u_hat slice (73.7 KB) lives
// in LDS (<< 320 KB WGP LDS); all softmax/reductions/squash in LDS.
// ---------------------------------------------------------------------------
__global__ __launch_bounds__(256) void routing_kernel(
    const float* __restrict__ uh_g,   // u_hat region, (C*B, N, 16)
    float* __restrict__ out)          // d_out base
{
    extern __shared__ float smem[];
    float* uh   = smem;                  // N*16 = 18432 floats
    float* abuf = uh + N_DIM * DOUT;     // 1152  (b_ij logits, scalar per n)
    float* red  = abuf + N_DIM;          // 256   (block reduction)
    float* sj   = red + 256;             // 16
    float* vj   = sj + 16;               // 16

    const int tid = threadIdx.x;
    const int cb  = blockIdx.x;          // c*B + b
    const float* src = uh_g + (size_t)cb * (N_DIM * DOUT);

    // Stage u_hat slice into LDS (contiguous, float4)
    {
        const v4f* s4 = (const v4f*)src;
        v4f* d4 = (v4f*)uh;
        for (int i = tid; i < N_DIM * DOUT / 4; i += 256) d4[i] = s4[i];
    }
    if (tid < 16) sj[tid] = 0.f;
    __syncthreads();

    // Iter 0: c_ij uniform = 1/N -> s0 = mean_n u_hat, v0 = squash(s0)
    {
        float acc[16];
        #pragma unroll
        for (int o = 0; o < 16; ++o) acc[o] = 0.f;
        for (int nn = tid; nn < N_DIM; nn += 256) {
            const float* row = uh + nn * DOUT;
            #pragma unroll
            for (int o = 0; o < 16; ++o) acc[o] += row[o];
        }
        #pragma unroll
        for (int o = 0; o < 16; ++o) atomicAdd(&sj[o], acc[o]);
    }
    __syncthreads();
    if (tid == 0) {
        float sq = 0.f, s[16];
        #pragma unroll
        for (int o = 0; o < 16; ++o) { s[o] = sj[o] * (1.f / N_DIM); sq += s[o] * s[o]; }
        float scale = (sq / (1.f + sq)) * rsqrtf(sq);
        #pragma unroll
        for (int o = 0; o < 16; ++o) vj[o] = scale * s[o];
    }
    __syncthreads();

    // a0 = u_hat . v0  -> logits buffer
    for (int nn = tid; nn < N_DIM; nn += 256) {
        const float* row = uh + nn * DOUT;
        float d = 0.f;
        #pragma unroll
        for (int o = 0; o < 16; ++o) d += row[o] * vj[o];
        abuf[nn] = d;
    }
    __syncthreads();

    // Iterations 1 and 2: softmax over N, weighted sum, squash
    for (int it = 1; it <= 2; ++it) {
        // max_n
        float m = -INFINITY;
        for (int nn = tid; nn < N_DIM; nn += 256) m = fmaxf(m, abuf[nn]);
        red[tid] = m; __syncthreads();
        for (int st = 128; st > 0; st >>= 1) {
            if (tid < st) red[tid] = fmaxf(red[tid], red[tid + st]);
            __syncthreads();
        }
        m = red[0]; __syncthreads();
        // sum exp
        float e = 0.f;
        for (int nn = tid; nn < N_DIM; nn += 256) e += __expf(abuf[nn] - m);
        red[tid] = e; __syncthreads();
        for (int st = 128; st > 0; st >>= 1) {
            if (tid < st) red[tid] += red[tid + st];
            __syncthreads();
        }
        float invZ = 1.f / red[0];
        __syncthreads();
        // s = sum_n c_ij * u_hat
        if (tid < 16) sj[tid] = 0.f;
        __syncthreads();
        float acc[16];
        #pragma unroll
        for (int o = 0; o < 16; ++o) acc[o] = 0.f;
        for (int nn = tid; nn < N_DIM; nn += 256) {
            float cij = __expf(abuf[nn] - m) * invZ;
            const float* row = uh + nn * DOUT;
            #pragma unroll
            for (int o = 0; o < 16; ++o) acc[o] += cij * row[o];
        }
        #pragma unroll
        for (int o = 0; o < 16; ++o) atomicAdd(&sj[o], acc[o]);
        __syncthreads();
        if (tid == 0) {
            float sq = 0.f;
            #pragma unroll
            for (int o = 0; o < 16; ++o) sq += sj[o] * sj[o];
            float scale = (sq / (1.f + sq)) * rsqrtf(sq);
            #pragma unroll
            for (int o = 0; o < 16; ++o) vj[o] = scale * sj[o];
        }
        __syncthreads();

        if (it == 1) {
            // a1 = u_hat . v1 : this is the a_ij the reference returns
            for (int nn = tid; nn < N_DIM; nn += 256) {
                const float* row = uh + nn * DOUT;
                float d = 0.f;
                #pragma unroll
                for (int o = 0; o < 16; ++o) d += row[o] * vj[o];
                out[A_OFF + (size_t)cb * N_DIM + nn] = d;
                abuf[nn] += d;                 // b_ij update for iter 2
            }
            __syncthreads();
        } else {
            if (tid < 16) out[V_OFF + (size_t)cb * DOUT + tid] = vj[tid];
        }
    }
}

extern "C" void kernel_launch(void* const* d_in, const int* in_sizes, int n_in,
                              void* d_out, int out_size, void* d_ws, size_t ws_size,
                              hipStream_t stream) {
    const float* u = (const float*)d_in[0];   // (B,N,8)
    const float* W = (const float*)d_in[1];   // (C,N,8,16)
    float* out = (float*)d_out;

    // Kernel A: grid (C*16 b-tiles, N / (8 waves * 9 n)) = (160, 16), 8 waves/block
    dim3 gA(C_DIM * 16, N_DIM / (WAVES * NPW), 1);
    uhat_wmma_kernel<<<gA, 256, 0, stream>>>(u, W, out);

    // Kernel B: one block per (c,b); dynamic LDS ~79.5 KB
    size_t lds = (size_t)(N_DIM * DOUT + N_DIM + 256 + 16 + 16) * sizeof(float);
    routing_kernel<<<C_DIM * B_DIM, 256, lds, stream>>>(out + UH_OFF, out);
}